// DPhysics_29652454211779
// MI455X (gfx1250) — compile-verified
//
#include <hip/hip_runtime.h>

typedef __attribute__((ext_vector_type(2))) float v2f;
typedef __attribute__((ext_vector_type(8))) float v8f;

#define D_MAX   6.4f
#define GRID_RS 0.05f
#define MASS_C  40.0f
#define GRAV_C  9.81f
#define DT_C    0.01f
#define EPS_N   1e-6f

__device__ __forceinline__ float rk4_s(float x0, float d) {
    float k1 = DT_C * d;
    float k2 = DT_C * (d + 0.5f * k1);
    float k3 = DT_C * (d + 0.5f * k2);
    float k4 = DT_C * (d + k3);
    return x0 + (k1 + 2.0f * k2 + 2.0f * k3 + k4) / 6.0f;
}

__device__ __forceinline__ float wave_sum32(float v) {
#pragma unroll
    for (int off = 16; off > 0; off >>= 1) v += __shfl_down(v, off, 32);
    return v;
}

// ---------------------------------------------------------------------------
// Kernel 1: per-point physics, fused x_points RK4 output, 26 partial sums per
// block written to d_ws. 8-wave -> block reduction done with WMMA f32 16x16x4.
// ---------------------------------------------------------------------------
__global__ __launch_bounds__(256)
void phys_main(const float* __restrict__ x,   const float* __restrict__ xd,
               const float* __restrict__ omg,
               const float* __restrict__ xp,  const float* __restrict__ xdp,
               const float* __restrict__ zg,  const float* __restrict__ kg,
               const float* __restrict__ dg,  const float* __restrict__ mg,
               const int*   __restrict__ ml,  const int*   __restrict__ mr,
               float* __restrict__ out, float* __restrict__ partial,
               int N, int ppb, int W, int H, int stride)
{
    const int b    = blockIdx.x;
    const int base = blockIdx.y * ppb;
    const int tid  = threadIdx.x;
    const int lane = tid & 31;
    const int wave = tid >> 5;

    const float bx = x[3*b],   by = x[3*b+1],   bz = x[3*b+2];
    const float vx = xd[3*b],  vy = xd[3*b+1],  vz = xd[3*b+2];
    const float wx = omg[3*b], wy = omg[3*b+1], wz = omg[3*b+2];

    const size_t gsz = (size_t)H * W;
    const float* zb  = zg + (size_t)b * gsz;
    const float* kb  = kg + (size_t)b * gsz;
    const float* db  = dg + (size_t)b * gsz;
    const float* mb  = mg + (size_t)b * gsz;
    const float* xpb  = xp  + (size_t)b * N * 3;
    const float* xdpb = xdp + (size_t)b * N * 3;
    float* orow = out + (size_t)b * stride + 18;

    float acc[26];
#pragma unroll
    for (int q = 0; q < 26; ++q) acc[q] = 0.0f;

    for (int i = base + tid; i < base + ppb; i += 256) {
        // Speculative prefetch of next tile of point data (global_prefetch_b8)
        __builtin_prefetch(xpb  + (size_t)(i + 256) * 3, 0, 0);
        __builtin_prefetch(xdpb + (size_t)(i + 256) * 3, 0, 0);

        const float px = xpb[3*i],  py = xpb[3*i+1],  pz = xpb[3*i+2];
        const float qx = xdpb[3*i], qy = xdpb[3*i+1], qz = xdpb[3*i+2];

        // bilinear corner setup (matches jax: trunc-toward-zero then clip)
        float gx = (px + D_MAX) / GRID_RS;
        float gy = (py + D_MAX) / GRID_RS;
        int xi = (int)gx; xi = xi < 0 ? 0 : (xi > W - 2 ? W - 2 : xi);
        int yi = (int)gy; yi = yi < 0 ? 0 : (yi > H - 2 ? H - 2 : yi);
        float xf = gx - (float)xi;
        float yf = gy - (float)yi;
        int i00 = xi + W * yi;

        float z00 = zb[i00], z01 = zb[i00+W], z10 = zb[i00+1], z11 = zb[i00+W+1];
        float k00 = kb[i00], k01 = kb[i00+W], k10 = kb[i00+1], k11 = kb[i00+W+1];
        float d00 = db[i00], d01 = db[i00+W], d10 = db[i00+1], d11 = db[i00+W+1];
        float m00 = mb[i00], m01 = mb[i00+W], m10 = mb[i00+1], m11 = mb[i00+W+1];

        float omx = 1.0f - xf, omy = 1.0f - yf;
        float w00 = omx*omy, w01 = omx*yf, w10 = xf*omy, w11 = xf*yf;

        float zq  = z00*w00 + z01*w01 + z10*w10 + z11*w11;
        float kq  = k00*w00 + k01*w01 + k10*w10 + k11*w11;
        float dq  = d00*w00 + d01*w01 + d10*w10 + d11*w11;
        float muq = m00*w00 + m01*w01 + m10*w10 + m11*w11;

        // surface normal
        float dzdx = ((z10 - z00) * omy + (z11 - z01) * yf) / GRID_RS;
        float dzdy = ((z01 - z00) * omx + (z11 - z10) * xf) / GRID_RS;
        float nx = -dzdx, ny = -dzdy;
        float nrm  = sqrtf(nx*nx + ny*ny + 1.0f);
        float rinv = 1.0f / (nrm + EPS_N);
        nx *= rinv; ny *= rinv;
        float nz = rinv;  // 1 * rinv

        float dh = pz - zq;
        bool ong = (px >= -D_MAX) && (px <= D_MAX) && (py >= -D_MAX) && (py <= D_MAX);
        float contact = (dh <= 0.0f && ong) ? 1.0f : 0.0f;

        float xdn  = qx*nx + qy*ny + qz*nz;
        float fmag = -(kq * dh + dq * xdn) * contact;
        float fsx = fmag*nx, fsy = fmag*ny, fsz = fmag*nz;
        float Nmag = sqrtf(fsx*fsx + fsy*fsy + fsz*fsz);

        float tx = qx - xdn*nx, ty = qy - xdn*ny, tz = qz - xdn*nz;
        float tnm  = sqrtf(tx*tx + ty*ty + tz*tz);
        float ffc  = -muq * Nmag / (tnm + EPS_N);
        float ffx = ffc*tx, ffy = ffc*ty, ffz = ffc*tz;

        float rx = px - bx, ry = py - by, rz = pz - bz;
        float fx = fsx + ffx, fy = fsy + ffy, fz = fsz + ffz;
        float tqx = ry*fz - rz*fy;
        float tqy = rz*fx - rx*fz;
        float tqz = rx*fy - ry*fx;

        float wl = (ml[i] != 0) ? 1.0f : 0.0f;
        float wr = (mr[i] != 0) ? 1.0f : 0.0f;

        acc[0]+=fsx; acc[1]+=fsy; acc[2]+=fsz;
        acc[3]+=ffx; acc[4]+=ffy; acc[5]+=ffz;
        acc[6]+=tqx; acc[7]+=tqy; acc[8]+=tqz;
        acc[9]+=Nmag;
        acc[10]+=px*wl; acc[11]+=py*wl; acc[12]+=pz*wl;
        acc[13]+=qx*wl; acc[14]+=qy*wl; acc[15]+=qz*wl;
        acc[16]+=contact*wl; acc[17]+=wl;
        acc[18]+=px*wr; acc[19]+=py*wr; acc[20]+=pz*wr;
        acc[21]+=qx*wr; acc[22]+=qy*wr; acc[23]+=qz*wr;
        acc[24]+=contact*wr; acc[25]+=wr;

        // fused point integration: xd_points_new = xd + omega x (p - x); rk4
        float ndx = vx + (wy*rz - wz*ry);
        float ndy = vy + (wz*rx - wx*rz);
        float ndz = vz + (wx*ry - wy*rx);
        orow[3*i+0] = rk4_s(px, ndx);
        orow[3*i+1] = rk4_s(py, ndy);
        orow[3*i+2] = rk4_s(pz, ndz);
    }

    // ---- per-wave reduction ----
    __shared__ float Pw[8][32];
    __shared__ float Sred[32];
#pragma unroll
    for (int q = 0; q < 26; ++q) {
        float s = wave_sum32(acc[q]);
        if (lane == 0) Pw[wave][q] = s;
    }
    if (lane == 0) {
#pragma unroll
        for (int q = 26; q < 32; ++q) Pw[wave][q] = 0.0f;
    }
    __syncthreads();

    // ---- 8-wave -> block reduction, done on the matrix unit by wave 0 ----
#if __has_builtin(__builtin_amdgcn_wmma_f32_16x16x4_f32)
    if (tid < 32) {
        const int nn = lane & 15;
        const int kh = lane >> 4;           // which K half this lane supplies
        v2f a; a[0] = 1.0f; a[1] = 1.0f;    // A = ones(16x4) -> D rows = col sums
#pragma unroll
        for (int n0 = 0; n0 < 32; n0 += 16) {
            v8f c = {};
            v2f b0; b0[0] = Pw[2*kh + 0][n0 + nn]; b0[1] = Pw[2*kh + 1][n0 + nn];
            c = __builtin_amdgcn_wmma_f32_16x16x4_f32(
                    false, a, false, b0, (short)0, c, false, false);
            v2f b1; b1[0] = Pw[2*kh + 4][n0 + nn]; b1[1] = Pw[2*kh + 5][n0 + nn];
            c = __builtin_amdgcn_wmma_f32_16x16x4_f32(
                    false, a, false, b1, (short)0, c, false, false);
            if (lane < 16) Sred[n0 + lane] = c[0];   // D[0][n] = sum over 8 waves
        }
    }
#else
    if (tid < 26) {
        float s = 0.0f;
#pragma unroll
        for (int w = 0; w < 8; ++w) s += Pw[w][tid];
        Sred[tid] = s;
    }
#endif
    __syncthreads();

    if (tid < 26)
        partial[((size_t)b * gridDim.y + blockIdx.y) * 32 + tid] = Sred[tid];
}

// ---------------------------------------------------------------------------
// Kernel 2: deterministic sum of per-block partials, then serial rigid-body
// update + 18-float header write (one wave per batch).
// ---------------------------------------------------------------------------
__global__ __launch_bounds__(32)
void phys_final(const float* __restrict__ x,  const float* __restrict__ xd,
                const float* __restrict__ R,  const float* __restrict__ omg,
                const float* __restrict__ ul, const float* __restrict__ ur,
                const float* __restrict__ partial, float* __restrict__ out,
                int N, int split, int stride)
{
    const int b = blockIdx.x;
    const int t = threadIdx.x;
    __shared__ float S[26];
    if (t < 26) {
        float s = 0.0f;
        for (int k = 0; k < split; ++k)
            s += partial[((size_t)b * split + k) * 32 + t];
        S[t] = s;
    }
    __syncthreads();
    if (t != 0) return;

    const float X0 = x[3*b],  X1 = x[3*b+1],  X2 = x[3*b+2];
    const float V0 = xd[3*b], V1 = xd[3*b+1], V2 = xd[3*b+2];
    const float W0 = omg[3*b], W1 = omg[3*b+1], W2 = omg[3*b+2];
    float Rm[9];
#pragma unroll
    for (int j = 0; j < 9; ++j) Rm[j] = R[9*b + j];

    // thrust_dir = normalize(R[:, :, 0])
    float c0x = Rm[0], c0y = Rm[3], c0z = Rm[6];
    float tn = sqrtf(c0x*c0x + c0y*c0y + c0z*c0z);
    float ti = 1.0f / (tn + EPS_N);
    float thx = c0x*ti, thy = c0y*ti, thz = c0z*ti;

    float swl = S[17], swr = S[25];
    float xlx = S[10]/swl, xly = S[11]/swl, xlz = S[12]/swl;
    float dlx = S[13]/swl, dly = S[14]/swl, dlz = S[15]/swl;
    float cl  = S[16]/swl;
    float xrx = S[18]/swr, xry = S[19]/swr, xrz = S[20]/swr;
    float drx = S[21]/swr, dry = S[22]/swr, drz = S[23]/swr;
    float cr  = S[24]/swr;

    float vl = dlx*thx + dly*thy + dlz*thz;
    float vr = drx*thx + dry*thy + drz*thz;
    float Nmean = S[9] / (float)N;

    float fl = Nmean * (ul[b] - vl) * cl;
    float fr = Nmean * (ur[b] - vr) * cr;
    float Flx = fl*thx, Fly = fl*thy, Flz = fl*thz;
    float Frx = fr*thx, Fry = fr*thy, Frz = fr*thz;

    float alx = xlx - X0, aly = xly - X1, alz = xlz - X2;
    float arx = xrx - X0, ary = xry - X1, arz = xrz - X2;
    float ttx = (aly*Flz - alz*Fly) + (ary*Frz - arz*Fry);
    float tty = (alz*Flx - alx*Flz) + (arz*Frx - arx*Frz);
    float ttz = (alx*Fly - aly*Flx) + (arx*Fry - ary*Frx);

    float tqx = S[6] + ttx, tqy = S[7] + tty, tqz = S[8] + ttz;
    float od0 = 2.0f * tqx, od1 = tqy, od2 = tqz;   // torque @ diag(2,1,1)

    float invN = 1.0f / (float)N;
    float Fc0 = S[0]*invN + S[3]*invN + Flx + Frx;
    float Fc1 = S[1]*invN + S[4]*invN + Fly + Fry;
    float Fc2 = -MASS_C*GRAV_C + S[2]*invN + S[5]*invN + Flz + Frz;
    float invM = 1.0f / MASS_C;
    float xdd0 = Fc0*invM, xdd1 = Fc1*invM, xdd2 = Fc2*invM;

    float nv0 = rk4_s(V0, xdd0), nv1 = rk4_s(V1, xdd1), nv2 = rk4_s(V2, xdd2);
    float nx0 = rk4_s(X0, nv0),  nx1 = rk4_s(X1, nv1),  nx2 = rk4_s(X2, nv2);
    float no0 = rk4_s(W0, od0),  no1 = rk4_s(W1, od1),  no2 = rk4_s(W2, od2);

    // rotation integration (Rodrigues with normalized skew)
    float wn = sqrtf(no0*no0 + no1*no1 + no2*no2);
    float theta = wn * DT_C;
    float dinv = 1.0f / (theta / DT_C + EPS_N);
    float sth = sinf(theta);
    float cth = 1.0f - cosf(theta);
    float O[9] = { 0.0f,      -no2*dinv,  no1*dinv,
                   no2*dinv,   0.0f,     -no0*dinv,
                  -no1*dinv,   no0*dinv,  0.0f };
    float O2[9];
#pragma unroll
    for (int i = 0; i < 3; ++i)
#pragma unroll
        for (int j = 0; j < 3; ++j)
            O2[3*i+j] = O[3*i]*O[j] + O[3*i+1]*O[3+j] + O[3*i+2]*O[6+j];
    float Mt[9];
#pragma unroll
    for (int idx = 0; idx < 9; ++idx)
        Mt[idx] = ((idx == 0 || idx == 4 || idx == 8) ? 1.0f : 0.0f)
                  + O[idx]*sth + O2[idx]*cth;
    float Rn[9];
#pragma unroll
    for (int i = 0; i < 3; ++i)
#pragma unroll
        for (int j = 0; j < 3; ++j)
            Rn[3*i+j] = Rm[3*i]*Mt[j] + Rm[3*i+1]*Mt[3+j] + Rm[3*i+2]*Mt[6+j];

    float* o = out + (size_t)b * stride;
    o[0] = nx0; o[1] = nx1; o[2] = nx2;
    o[3] = nv0; o[4] = nv1; o[5] = nv2;
#pragma unroll
    for (int j = 0; j < 9; ++j) o[6+j] = Rn[j];
    o[15] = no0; o[16] = no1; o[17] = no2;
}

extern "C" void kernel_launch(void* const* d_in, const int* in_sizes, int n_in,
                              void* d_out, int out_size, void* d_ws, size_t ws_size,
                              hipStream_t stream) {
    const float* x   = (const float*)d_in[0];
    const float* xd  = (const float*)d_in[1];
    const float* R   = (const float*)d_in[2];
    const float* omg = (const float*)d_in[3];
    const float* xp  = (const float*)d_in[4];
    const float* xdp = (const float*)d_in[5];
    const float* zg  = (const float*)d_in[6];
    const float* kg  = (const float*)d_in[7];
    const float* dg  = (const float*)d_in[8];
    const float* mg  = (const float*)d_in[9];
    const float* ul  = (const float*)d_in[10];
    const float* ur  = (const float*)d_in[11];
    const int*   ml  = (const int*)d_in[12];
    const int*   mr  = (const int*)d_in[13];
    float* out     = (float*)d_out;
    float* partial = (float*)d_ws;

    const int B  = in_sizes[0] / 3;
    const int N  = in_sizes[12];
    const int HW = in_sizes[6] / B;
    int W = 1;
    while ((long long)(W + 1) * (W + 1) <= (long long)HW) ++W;  // integer sqrt
    const int H = HW / W;
    const int stride = out_size / B;

    int split = 8;
    while (split > 1 &&
           ((N % (split * 256)) != 0 ||
            (size_t)B * split * 32 * sizeof(float) > ws_size))
        split >>= 1;
    const int ppb = N / split;

    dim3 grid(B, split);
    phys_main<<<grid, 256, 0, stream>>>(x, xd, omg, xp, xdp, zg, kg, dg, mg,
                                        ml, mr, out, partial, N, ppb, W, H, stride);
    phys_final<<<B, 32, 0, stream>>>(x, xd, R, omg, ul, ur, partial, out,
                                     N, split, stride);
}